// L2DPopModel_55001351193211
// MI455X (gfx1250) — compile-verified
//
#include <hip/hip_runtime.h>

// ---------------------------------------------------------------------------
// L2D PopModel forward, algebraically simplified for MI455X (gfx1250, wave32)
//   features = x @ Wf + bf                       (512 x 1024, K=2048)
//   clf      = features @ Wc + bc                (512 x 100,  K=1024)
//   A        = features @ W1[:F]                 (512 x 768,  K=1024)
//   Hc       = human_rep @ W1[F:]                (64  x 768,  K=1536)
//   BN over (B,H): mean/var separable -> s_k, tA_k, tH_k  (b1 cancels)
//   out[b,j,100] = sum_k relu(A'[b,k] + H''[j,k]) * W2[k] + b2
//   out[b,j,0:100] = clf[b,:]
// GEMMs use V_WMMA_F32_16X16X4_F32 (fp32-exact CDNA5 matrix path).
// ---------------------------------------------------------------------------

typedef __attribute__((ext_vector_type(2))) float v2f;
typedef __attribute__((ext_vector_type(8))) float v8f;

#define NB   512
#define NH   64
#define NF   1024
#define NE   512
#define NC   100
#define NHID 768
#define NDIN 2048
#define WPB  4          // waves per block in GEMM kernel

// Generic fp32 WMMA GEMM:  D = A(MxK) * B(KxN) (+bias).  Row-major.
// Each wave computes a 32x64 strip of D: 2 M-tiles x 4 N-tiles of 16x16,
// so each B fragment feeds two WMMAs (10 loads -> 8 WMMAs per K-step).
// K % 4 == 0; M % 32 == 0; N arbitrary when GUARD (mask-folded, branchless).
template <bool GUARD>
__global__ __launch_bounds__(128) void gemm_f32_wmma(
    const float* __restrict__ A, const float* __restrict__ B,
    const float* __restrict__ bias, float* __restrict__ D,
    int M, int N, int K, int lda, int ldb, int ldd)
{
    const int lane = threadIdx.x & 31;
    const int wv   = threadIdx.x >> 5;
    const int wid  = blockIdx.x * WPB + wv;

    const int mgrp = M >> 5;             // 32-row groups
    const int nreg = (N + 63) >> 6;      // 64-col groups
    const int rm   = wid / nreg;
    const int rn   = wid % nreg;
    if (rm >= mgrp) return;              // wave-uniform exit

    const int m    = lane & 15;          // row within 16-tile (also col-in-tile)
    const int half = lane >> 4;          // 0 or 1
    const int krow = half << 1;          // K offset {0,2} for this lane half

    const float* Arow0 = A + (size_t)((rm << 5) + m) * lda;
    const float* Arow1 = Arow0 + (size_t)16 * lda;
    const int    ncol0 = (rn << 6) + m;  // first of 4 column positions

    // Loop-invariant column clamp + 0/1 mask: keeps the K loop branchless.
    int   colt[4];
    float maskt[4];
#pragma unroll
    for (int t = 0; t < 4; ++t) {
        const int c = ncol0 + (t << 4);
        colt[t]  = (!GUARD || c < N) ? c : 0;
        maskt[t] = (!GUARD || c < N) ? 1.0f : 0.0f;
    }

    v8f acc[2][4] = {};

    for (int k = 0; k < K; k += 4) {
        // gfx1250 prefetch path (global_prefetch_b8): A 16 K-steps (64 k) ahead, B 4 K-steps (16 k) ahead
        __builtin_prefetch(Arow0 + k + 64, 0, 3);
        __builtin_prefetch(B + (size_t)(k + krow + 16) * ldb + ncol0, 0, 3);

        // A fragments: lanes 0-15 hold K=k,k+1 ; lanes 16-31 hold K=k+2,k+3
        v2f a0 = *(const v2f*)(Arow0 + k + krow);
        v2f a1 = *(const v2f*)(Arow1 + k + krow);

        const float* Bk0 = B + (size_t)(k + krow) * ldb;
        const float* Bk1 = Bk0 + ldb;
        v2f bf[4];
#pragma unroll
        for (int t = 0; t < 4; ++t) {
            if (GUARD) {
                bf[t].x = Bk0[colt[t]] * maskt[t];
                bf[t].y = Bk1[colt[t]] * maskt[t];
            } else {
                const int col = ncol0 + (t << 4);
                bf[t].x = Bk0[col];
                bf[t].y = Bk1[col];
            }
        }
#pragma unroll
        for (int t = 0; t < 4; ++t) {
            acc[0][t] = __builtin_amdgcn_wmma_f32_16x16x4_f32(
                false, a0, false, bf[t], (short)0, acc[0][t], false, false);
            acc[1][t] = __builtin_amdgcn_wmma_f32_16x16x4_f32(
                false, a1, false, bf[t], (short)0, acc[1][t], false, false);
        }
    }

    // Epilogue: VGPR r of C/D holds rows {r, r+8} split across lane halves.
#pragma unroll
    for (int mi = 0; mi < 2; ++mi) {
#pragma unroll
        for (int t = 0; t < 4; ++t) {
            const int col = ncol0 + (t << 4);
            if (!GUARD || col < N) {
                const float bv = bias ? bias[col] : 0.0f;
#pragma unroll
                for (int r = 0; r < 8; ++r) {
                    const int row = (rm << 5) + (mi << 4) + r + (half << 3);
                    D[(size_t)row * ldd + col] = acc[mi][t][r] + bv;
                }
            }
        }
    }
}

// Per-feature BN statistics over the separable decomposition.
// var_(b,j)(A_b + H_j + b1) = Var_b(A) + Var_j(H); b1 cancels in (h - mean).
__global__ __launch_bounds__(256) void bn_stats(
    const float* __restrict__ Af, const float* __restrict__ Hc,
    const float* __restrict__ bns, const float* __restrict__ bnb,
    float* __restrict__ sK, float* __restrict__ tA, float* __restrict__ tH)
{
    const int k = blockIdx.x * 256 + threadIdx.x;
    if (k >= NHID) return;
    float sum = 0.f, sq = 0.f;
    for (int b = 0; b < NB; ++b) {
        float v = Af[(size_t)b * NHID + k];
        sum += v; sq += v * v;
    }
    const float mA = sum * (1.0f / NB);
    const float vA = sq  * (1.0f / NB) - mA * mA;
    sum = 0.f; sq = 0.f;
    for (int j = 0; j < NH; ++j) {
        float v = Hc[j * NHID + k];
        sum += v; sq += v * v;
    }
    const float mH = sum * (1.0f / NH);
    const float vH = sq  * (1.0f / NH) - mH * mH;
    const float s  = bns[k] * rsqrtf(vA + vH + 1e-5f);
    sK[k] = s;
    tA[k] = -mA * s;
    tH[k] = bnb[k] - mH * s;
}

// H''[j,k] = Hc[j,k] * s_k + tH_k   (64 x 768)
__global__ __launch_bounds__(256) void apply_h(
    const float* __restrict__ Hc, const float* __restrict__ sK,
    const float* __restrict__ tH, float* __restrict__ Hpp)
{
    const int idx = blockIdx.x * 256 + threadIdx.x;
    if (idx >= NH * NHID) return;
    const int k = idx % NHID;
    Hpp[idx] = Hc[idx] * sK[k] + tH[k];
}

// Fused epilogue: one block per batch row b. Stage transformed A-row + W2 in
// LDS; 8 wave32 waves sweep the 64 human slots (ReLU + dot with W2, shfl_xor
// reduction); broadcast classifier logits into the (B,H,101) output.
__global__ __launch_bounds__(256) void fuse_out(
    const float* __restrict__ Af, const float* __restrict__ Hpp,
    const float* __restrict__ sK, const float* __restrict__ tA,
    const float* __restrict__ clf, const float* __restrict__ W2,
    const float* __restrict__ b2, float* __restrict__ out)
{
    __shared__ float Ar[NHID];
    __shared__ float w2s[NHID];
    __shared__ float clfs[NC];

    const int b   = blockIdx.x;
    const int tid = threadIdx.x;

    for (int k = tid; k < NHID; k += 256) {
        Ar[k]  = Af[(size_t)b * NHID + k] * sK[k] + tA[k];
        w2s[k] = W2[k];
    }
    for (int i = tid; i < NC; i += 256) clfs[i] = clf[(size_t)b * NC + i];
    __syncthreads();

    const int wv   = tid >> 5;
    const int lane = tid & 31;
    const float bias2 = b2[0];

#pragma unroll
    for (int jj = 0; jj < NH / 8; ++jj) {
        const int j = wv + (jj << 3);
        const float* Hrow = Hpp + j * NHID;
        float sum = 0.f;
        for (int k = lane; k < NHID; k += 32) {
            const float v = Ar[k] + Hrow[k];
            sum += fmaxf(v, 0.f) * w2s[k];
        }
#pragma unroll
        for (int off = 16; off > 0; off >>= 1)
            sum += __shfl_xor(sum, off, 32);
        if (lane == 0)
            out[((size_t)b * NH + j) * (NC + 1) + NC] = sum + bias2;
    }

    for (int idx = tid; idx < NH * NC; idx += 256) {
        const int j = idx / NC;
        const int c = idx - j * NC;
        out[((size_t)b * NH + j) * (NC + 1) + c] = clfs[c];
    }
}

extern "C" void kernel_launch(void* const* d_in, const int* in_sizes, int n_in,
                              void* d_out, int out_size, void* d_ws, size_t ws_size,
                              hipStream_t stream) {
    const float* x   = (const float*)d_in[0];
    const float* hr  = (const float*)d_in[1];
    const float* Wf  = (const float*)d_in[2];
    const float* bf  = (const float*)d_in[3];
    const float* Wc  = (const float*)d_in[4];
    const float* bc  = (const float*)d_in[5];
    const float* W1  = (const float*)d_in[6];
    // d_in[7] = b1 : cancels exactly in training-mode BatchNorm
    const float* bns = (const float*)d_in[8];
    const float* bnb = (const float*)d_in[9];
    const float* W2  = (const float*)d_in[10];
    const float* b2  = (const float*)d_in[11];
    float* out = (float*)d_out;

    float* ws   = (float*)d_ws;
    float* feat = ws;                         // 512*1024
    float* Af   = feat + (size_t)NB * NF;     // 512*768
    float* Hc   = Af   + (size_t)NB * NHID;   // 64*768
    float* clf  = Hc   + (size_t)NH * NHID;   // 512*100
    float* sK   = clf  + (size_t)NB * NC;     // 768
    float* tA   = sK + NHID;                  // 768
    float* tH   = tA + NHID;                  // 768
    float* Hpp  = tH + NHID;                  // 64*768

    auto gemm_full = [&](const float* Am, const float* Bm, const float* bias,
                         float* Dm, int M, int N, int K, int lda, int ldb, int ldd) {
        const int regions = (M >> 5) * ((N + 63) >> 6);
        const int blocks  = (regions + WPB - 1) / WPB;
        gemm_f32_wmma<false><<<blocks, 128, 0, stream>>>(Am, Bm, bias, Dm,
                                                         M, N, K, lda, ldb, ldd);
    };
    auto gemm_guard = [&](const float* Am, const float* Bm, const float* bias,
                          float* Dm, int M, int N, int K, int lda, int ldb, int ldd) {
        const int regions = (M >> 5) * ((N + 63) >> 6);
        const int blocks  = (regions + WPB - 1) / WPB;
        gemm_f32_wmma<true><<<blocks, 128, 0, stream>>>(Am, Bm, bias, Dm,
                                                        M, N, K, lda, ldb, ldd);
    };

    // features = x @ Wf + bf
    gemm_full(x, Wf, bf, feat, NB, NF, NDIN, NDIN, NF, NF);
    // A = features @ W1[:F, :]
    gemm_full(feat, W1, nullptr, Af, NB, NHID, NF, NF, NHID, NHID);
    // Hc = human_rep @ W1[F:, :]
    gemm_full(hr, W1 + (size_t)NF * NHID, nullptr, Hc, NH, NHID, NF + NE,
              NF + NE, NHID, NHID);
    // clf = features @ Wc + bc  (N=100 -> masked columns)
    gemm_guard(feat, Wc, bc, clf, NB, NC, NF, NF, NC, NC);

    bn_stats<<<(NHID + 255) / 256, 256, 0, stream>>>(Af, Hc, bns, bnb, sK, tA, tH);
    apply_h<<<(NH * NHID + 255) / 256, 256, 0, stream>>>(Hc, sK, tH, Hpp);
    fuse_out<<<NB, 256, 0, stream>>>(Af, Hpp, sK, tA, clf, W2, b2, out);
}